// CPDecoding_69423851372726
// MI455X (gfx1250) — compile-verified
//
#include <hip/hip_runtime.h>
#include <hip/hip_fp16.h>
#include <stdint.h>

// ---------------------------------------------------------------------------
// CP decoding on MI455X (gfx1250):
//   out[n] = sum_c  lerp(line_z[c], z_n) * lerp(line_y[c], y_n) * lerp(line_x[c], x_n)
//
//   prep:       (96,512) f32 tables -> (512,96) f16 transposed tables in d_ws.
//   cp_decode:  async-stage all three tables (288 KB payload) into LDS with
//               global_load_async_to_lds_b128 (ASYNCcnt), padding rows
//               192B -> 208B (52 dwords, 16B-aligned => ds_load_b128, random
//               row conflicts only when i0 = i0' mod 16); then one point per
//               lane: packed-f16 lerps/products (v_pk_fma_f16), f32 accumulate.
// ---------------------------------------------------------------------------

#define NC        96                      // components
#define RES       512                     // resolution per axis
#define ROW_B     192                     // bytes per row in d_ws (96 * 2B)
#define ROW_LDS_B 208                     // bytes per row in LDS (16B aligned)
#define RS_H      104                     // LDS row stride in halves
#define TBL_H     (RES * RS_H)            // halves per table in LDS (53248)
#define LDS_BYTES (3 * RES * ROW_LDS_B)   // 319488 B (<= 320 KB WGP LDS)
#define WS_HALVES (3 * RES * NC)          // 147456 halves in d_ws (288 KB)

typedef __attribute__((ext_vector_type(4))) unsigned int uint4v;  // 16B LDS chunk
union V16 { uint4v u; __half2 h[4]; };

// ---------- prep: transpose + f32->f16 convert into workspace --------------
__global__ void cp_prep_kernel(const float* __restrict__ lz,
                               const float* __restrict__ ly,
                               const float* __restrict__ lx,
                               __half* __restrict__ wsT) {
  int t = blockIdx.x * blockDim.x + threadIdx.x;
  if (t >= WS_HALVES) return;
  int table = t / (NC * RES);
  int rem   = t - table * (NC * RES);
  int c     = rem / RES;          // component
  int i     = rem - c * RES;      // grid index (fastest -> coalesced reads)
  const float* src = (table == 0) ? lz : (table == 1) ? ly : lx;
  wsT[table * (RES * NC) + i * NC + c] = __float2half(src[c * RES + i]);
}

// ---------- index/weight computation (matches reference exactly) -----------
__device__ __forceinline__ void cp_idx(float coord, int& i0, int& i1, float& w) {
  float pos = (coord + 1.0f) * 0.5f * (float)(RES - 1);   // (c+1)*255.5
  float f   = floorf(pos);
  f  = fminf(fmaxf(f, 0.0f), (float)(RES - 1));
  i0 = (int)f;
  i1 = min(i0 + 1, RES - 1);
  w  = pos - f;
}

// ---------- main kernel ----------------------------------------------------
__global__ void __launch_bounds__(512)
cp_decode_kernel(const float* __restrict__ pts,     // (N,3)
                 const __half* __restrict__ wsT,    // 3*(512,96) f16
                 float* __restrict__ out, int N) {
  extern __shared__ __align__(16) __half smem[];

  // ---- async-stage all three tables into LDS, restriding 192B -> 208B rows
  {
    const uint32_t lds_base = (uint32_t)(uintptr_t)(&smem[0]);
    const uint64_t g_base   = (uint64_t)(uintptr_t)wsT;
    const int nchunks = 3 * RES * (ROW_B / 16);   // 18432 b128 chunks
    for (int k = threadIdx.x; k < nchunks; k += blockDim.x) {
      int r = k / (ROW_B / 16);                   // global row (table*512 + i)
      int j = k - r * (ROW_B / 16);               // 16-byte chunk within row
      uint32_t lds_addr = lds_base + (uint32_t)(r * ROW_LDS_B + j * 16);
      uint64_t gaddr    = g_base   + (uint64_t)(r * ROW_B     + j * 16);
      asm volatile("global_load_async_to_lds_b128 %0, %1, off"
                   :: "v"(lds_addr), "v"(gaddr) : "memory");
    }
    asm volatile("s_wait_asynccnt 0x0" ::: "memory");
  }
  __syncthreads();

  const __half* Sz = smem;               // table z
  const __half* Sy = smem + TBL_H;       // table y
  const __half* Sx = smem + 2 * TBL_H;   // table x

  const int tid    = blockIdx.x * blockDim.x + threadIdx.x;
  const int stride = gridDim.x * blockDim.x;

  for (int n = tid; n < N; n += stride) {
    float px = pts[3 * n + 0];
    float py = pts[3 * n + 1];
    float pz = pts[3 * n + 2];

    int iz0, iz1, iy0, iy1, ix0, ix1;
    float wz, wy, wx;
    cp_idx(pz, iz0, iz1, wz);   // fz samples pts[:,2] against line_z
    cp_idx(py, iy0, iy1, wy);
    cp_idx(px, ix0, ix1, wx);

    const __half2 wz2 = __float2half2_rn(wz);
    const __half2 wy2 = __float2half2_rn(wy);
    const __half2 wx2 = __float2half2_rn(wx);

    // 16B-aligned row starts (RS_H*2 = 208 = 13*16)
    const uint4v* z0 = (const uint4v*)(Sz + iz0 * RS_H);
    const uint4v* z1 = (const uint4v*)(Sz + iz1 * RS_H);
    const uint4v* y0 = (const uint4v*)(Sy + iy0 * RS_H);
    const uint4v* y1 = (const uint4v*)(Sy + iy1 * RS_H);
    const uint4v* x0 = (const uint4v*)(Sx + ix0 * RS_H);
    const uint4v* x1 = (const uint4v*)(Sx + ix1 * RS_H);

    float acc0 = 0.0f, acc1 = 0.0f;
#pragma unroll 4
    for (int j = 0; j < NC / 8; ++j) {    // 12 blocks of 8 components
      V16 A0, A1, B0, B1, C0, C1;
      A0.u = z0[j]; A1.u = z1[j];
      B0.u = y0[j]; B1.u = y1[j];
      C0.u = x0[j]; C1.u = x1[j];
#pragma unroll
      for (int q = 0; q < 4; ++q) {       // 4 x half2 per 16B chunk
        __half2 fz = __hfma2(wz2, __hsub2(A1.h[q], A0.h[q]), A0.h[q]);
        __half2 fy = __hfma2(wy2, __hsub2(B1.h[q], B0.h[q]), B0.h[q]);
        __half2 fx = __hfma2(wx2, __hsub2(C1.h[q], C0.h[q]), C0.h[q]);
        __half2 p  = __hmul2(fz, __hmul2(fy, fx));
        float2  f  = __half22float2(p);   // fused into v_fma_mix_f32 adds
        acc0 += f.x;
        acc1 += f.y;
      }
    }
    out[n] = acc0 + acc1;
  }
}

// ---------------------------------------------------------------------------
extern "C" void kernel_launch(void* const* d_in, const int* in_sizes, int n_in,
                              void* d_out, int out_size, void* d_ws, size_t ws_size,
                              hipStream_t stream) {
  const float* in_tensor = (const float*)d_in[0];   // (4096,192,3)
  const float* line_z    = (const float*)d_in[1];   // (96,512)
  const float* line_y    = (const float*)d_in[2];
  const float* line_x    = (const float*)d_in[3];
  float*       out       = (float*)d_out;
  __half*      wsT       = (__half*)d_ws;           // 288 KB used

  const int N = in_sizes[0] / 3;                    // 786432 points

  // 1) transpose + convert tables into workspace
  {
    int total = WS_HALVES;
    int block = 256;
    int grid  = (total + block - 1) / block;
    cp_prep_kernel<<<grid, block, 0, stream>>>(line_z, line_y, line_x, wsT);
  }

  // 2) decode; one workgroup per WGP (LDS-bound), grid-stride over points
  {
    dim3 block(512);          // 16 wave32 waves
    dim3 grid(384);           // ~4 points per thread
    cp_decode_kernel<<<grid, block, LDS_BYTES, stream>>>(in_tensor, wsT, out, N);
  }
}